// DS_EmbeddingGenerator_69758858821831
// MI455X (gfx1250) — compile-verified
//
#include <hip/hip_runtime.h>

#define L_  4
#define N_  200000
#define E_  3200000
#define F_  20
#define H_  32
#define D_  32
#define LH_ 64

typedef __attribute__((ext_vector_type(2))) float v2f;
typedef __attribute__((ext_vector_type(8))) float v8f;

// ---------------------------------------------------------------------------
// WMMA f32 GEMM: C[M, NT*16] = A[M,K] @ B[K, NT*16] (+bias) (+relu)
// One wave per 16-row strip; NT 16x16 column tiles held in v8f accumulators.
// Uses V_WMMA_F32_16X16X4_F32 (full fp32 precision, K stepped by 4).
// Fragment mapping per CDNA5 ISA 7.12.2:
//   A 16x4 : lane&15 = M row; VGPR j holds K = k + 2*(lane>>4) + j
//   B 4x16 : lane&15 = N col; VGPR j holds K = k + 2*(lane>>4) + j
//   C 16x16: lane&15 = N col; VGPR r holds M = rowBase + r + 8*(lane>>4)
// ---------------------------------------------------------------------------
template <int NT>
__global__ void gemm_wmma_f32(const float* __restrict__ A,
                              const float* __restrict__ B,
                              const float* __restrict__ bias,
                              float* __restrict__ C,
                              int M, int K, int doRelu) {
  constexpr int Nc = NT * 16;
  const int wave    = blockIdx.x * (blockDim.x >> 5) + (threadIdx.x >> 5);
  const int lane    = threadIdx.x & 31;
  const int rowBase = wave * 16;
  if (rowBase >= M) return;                 // wave-uniform: EXEC stays all-ones
  const int lhalf = lane >> 4;
  const int l16   = lane & 15;

  v8f acc[NT] = {};

  int arow = rowBase + l16;
  if (arow >= M) arow = M - 1;
  const float* Ar = A + (size_t)arow * K;

  for (int k = 0; k < K; k += 4) {
    const int kk = k + 2 * lhalf;
    v2f a;
    a.x = Ar[kk];
    a.y = Ar[kk + 1];
#pragma unroll
    for (int t = 0; t < NT; ++t) {
      const int col = t * 16 + l16;
      v2f b;
      b.x = B[(size_t)kk * Nc + col];
      b.y = B[(size_t)(kk + 1) * Nc + col];
      acc[t] = __builtin_amdgcn_wmma_f32_16x16x4_f32(
          /*neg_a=*/false, a, /*neg_b=*/false, b,
          /*c_mod=*/(short)0, acc[t], /*reuse_a=*/false, /*reuse_b=*/false);
    }
  }

#pragma unroll
  for (int t = 0; t < NT; ++t) {
    const int col = t * 16 + l16;
    const float bv = bias ? bias[col] : 0.0f;
#pragma unroll
    for (int r = 0; r < 8; ++r) {
      const int row = rowBase + r + 8 * lhalf;
      if (row < M) {
        float v = acc[t][r] + bv;
        if (doRelu) v = fmaxf(v, 0.0f);
        C[(size_t)row * Nc + col] = v;
      }
    }
  }
}

// ---------------------------------------------------------------------------
// Elementwise / scatter kernels (C == 32 channels == one wave32 per edge)
// ---------------------------------------------------------------------------
__global__ void k_fill_zero(float* __restrict__ p, int n) {
  int i = blockIdx.x * blockDim.x + threadIdx.x;
  if (i < n) p[i] = 0.0f;
}

__global__ void k_deg_count(const int* __restrict__ dst, float* __restrict__ deg, int E) {
  int e = blockIdx.x * blockDim.x + threadIdx.x;
  if (e < E) atomicAdd(&deg[dst[e]], 1.0f);
}

__global__ void k_inv(float* __restrict__ d, int n) {
  int i = blockIdx.x * blockDim.x + threadIdx.x;
  if (i < n) d[i] = rsqrtf(d[i] + 1.0f);
}

// out[n,c] = hw[n,c] * inv[n]^2 + b[c]   (self-loop term + bias, also init)
__global__ void k_self_init(const float* __restrict__ hw, const float* __restrict__ inv,
                            const float* __restrict__ b, float* __restrict__ out, int n) {
  int i = blockIdx.x * blockDim.x + threadIdx.x;
  if (i < n) {
    int node = i >> 5, c = i & 31;
    float iv = inv[node];
    out[i] = hw[i] * iv * iv + b[c];
  }
}

// out[dst,c] += hw[src,c] * inv[src]*inv[dst]; one wave per edge, lane = chan
__global__ void k_edge_agg(const int* __restrict__ src, const int* __restrict__ dst,
                           const float* __restrict__ inv, const float* __restrict__ hw,
                           float* __restrict__ out, int EC) {
  int t = blockIdx.x * blockDim.x + threadIdx.x;
  if (t >= EC) return;
  int e = t >> 5, c = t & 31;
  int s = src[e], d = dst[e];
  float coef = inv[s] * inv[d];
  atomicAdd(&out[(size_t)d * 32 + c], hw[(size_t)s * 32 + c] * coef);
}

__global__ void k_relu(float* __restrict__ p, int n) {
  int i = blockIdx.x * blockDim.x + threadIdx.x;
  if (i < n) p[i] = fmaxf(p[i], 0.0f);
}

// acc[recv,c] += last[srcL,c]
__global__ void k_msg_agg(const int* __restrict__ recv, const int* __restrict__ srcL,
                          const float* __restrict__ last, float* __restrict__ acc, int EC) {
  int t = blockIdx.x * blockDim.x + threadIdx.x;
  if (t >= EC) return;
  int e = t >> 5, c = t & 31;
  atomicAdd(&acc[(size_t)recv[e] * 32 + c], last[(size_t)srcL[e] * 32 + c]);
}

// last = (feat + acc) / deg_scalars[layer]
__global__ void k_combine(const float* __restrict__ feat, const float* __restrict__ acc,
                          const float* __restrict__ degs, int layer,
                          float* __restrict__ last, int n) {
  int i = blockIdx.x * blockDim.x + threadIdx.x;
  if (i < n) last[i] = (feat[i] + acc[i]) / degs[layer];
}

// ---------------------------------------------------------------------------
extern "C" void kernel_launch(void* const* d_in, const int* in_sizes, int n_in,
                              void* d_out, int out_size, void* d_ws, size_t ws_size,
                              hipStream_t stream) {
  const float* xs  = (const float*)d_in[0];   // [L,N,F]
  const int*   ei  = (const int*)  d_in[1];   // [L,2,E]  row0=dst, row1=src
  const int*   lei = (const int*)  d_in[2];   // [L,2,E]  row0=recv, row1=src
  const float* dsc = (const float*)d_in[3];   // [L]
  const float* gw1 = (const float*)d_in[4];   // [L,F,H]
  const float* gb1 = (const float*)d_in[5];   // [L,H]
  const float* gw2 = (const float*)d_in[6];   // [L,H,D]
  const float* gb2 = (const float*)d_in[7];   // [L,D]
  const float* l1w = (const float*)d_in[8];   // [D,LH]
  const float* l1b = (const float*)d_in[9];   // [LH]
  const float* l2w = (const float*)d_in[10];  // [LH,D]
  const float* l2b = (const float*)d_in[11];  // [D]
  float* out = (float*)d_out;                 // [N,D]

  // workspace carve-out (~180 MB of f32)
  float* w    = (float*)d_ws;
  float* inv  = w;                            // N
  float* hw   = inv  + (size_t)N_;            // N*32
  float* hbuf = hw   + (size_t)N_ * H_;       // N*32
  float* feat = hbuf + (size_t)N_ * H_;       // N*32
  float* last = feat + (size_t)N_ * D_;       // N*32
  float* acc  = last + (size_t)N_ * D_;       // N*32
  float* hid  = acc  + (size_t)N_ * D_;       // N*64

  const int TPB = 256;
  auto nb = [](long n, int tpb) { return (int)((n + tpb - 1) / tpb); };
  const int gemmBlocks = (N_ / 16 + 3) / 4;   // 4 waves / 128-thread block

  for (int i = 0; i < L_; ++i) {
    const int* dst = ei + (size_t)i * 2 * E_;
    const int* src = dst + E_;

    // degrees -> inv = rsqrt(deg + 1)
    k_fill_zero<<<nb(N_, TPB), TPB, 0, stream>>>(inv, N_);
    k_deg_count<<<nb(E_, TPB), TPB, 0, stream>>>(dst, inv, E_);
    k_inv<<<nb(N_, TPB), TPB, 0, stream>>>(inv, N_);

    // conv1: hw = x @ w1 ; hbuf = agg + self + b1 ; relu
    gemm_wmma_f32<2><<<gemmBlocks, 128, 0, stream>>>(
        xs + (size_t)i * N_ * F_, gw1 + (size_t)i * F_ * H_, nullptr, hw, N_, F_, 0);
    k_self_init<<<nb((long)N_ * H_, TPB), TPB, 0, stream>>>(
        hw, inv, gb1 + (size_t)i * H_, hbuf, N_ * H_);
    k_edge_agg<<<nb((long)E_ * 32, TPB), TPB, 0, stream>>>(
        src, dst, inv, hw, hbuf, E_ * 32);
    k_relu<<<nb((long)N_ * H_, TPB), TPB, 0, stream>>>(hbuf, N_ * H_);

    // conv2: hw = h @ w2 ; conv2_out = agg + self + b2
    float* c2out = (i == 0) ? last : feat;    // layer 0 writes `last` directly
    gemm_wmma_f32<2><<<gemmBlocks, 128, 0, stream>>>(
        hbuf, gw2 + (size_t)i * H_ * D_, nullptr, hw, N_, H_, 0);
    k_self_init<<<nb((long)N_ * D_, TPB), TPB, 0, stream>>>(
        hw, inv, gb2 + (size_t)i * D_, c2out, N_ * D_);
    k_edge_agg<<<nb((long)E_ * 32, TPB), TPB, 0, stream>>>(
        src, dst, inv, hw, c2out, E_ * 32);

    // inter-layer message passing
    if (i > 0) {
      const int* recv = lei + (size_t)i * 2 * E_;
      const int* srcL = recv + E_;
      k_fill_zero<<<nb((long)N_ * D_, TPB), TPB, 0, stream>>>(acc, N_ * D_);
      k_msg_agg<<<nb((long)E_ * 32, TPB), TPB, 0, stream>>>(
          recv, srcL, last, acc, E_ * 32);
      k_combine<<<nb((long)N_ * D_, TPB), TPB, 0, stream>>>(
          feat, acc, dsc, i, last, N_ * D_);
    }
  }

  // final MLP: hid = relu(last @ l1w + l1b) ; out = relu(hid @ l2w + l2b)
  gemm_wmma_f32<4><<<gemmBlocks, 128, 0, stream>>>(last, l1w, l1b, hid, N_, D_, 1);
  gemm_wmma_f32<2><<<gemmBlocks, 128, 0, stream>>>(hid, l2w, l2b, out, N_, LH_, 1);
}